// ResBlock_21603685499560
// MI455X (gfx1250) — compile-verified
//
#include <hip/hip_runtime.h>

typedef float v2f __attribute__((ext_vector_type(2)));
typedef float v8f __attribute__((ext_vector_type(8)));

// LDS layout (floats). Row strides padded (+4 per row => 16 rows span 64 banks)
// so 16-lane column reads for WMMA A-fragments are bank-conflict free.
#define XSTR 964                 // x tile row stride (960 cols)
#define CSTR 708                 // cat0 row stride (704 cols)
#define GSTR 196                 // gates row stride (192 cols)
#define XB0  0
#define XB1  (16 * XSTR)
#define CATO (32 * XSTR)
#define GATO (CATO + 16 * CSTR)
#define SMEM_FLOATS (GATO + 16 * GSTR)   // 45312 floats = 181248 bytes < 320KB

// scales (constant-folded)
#define INV_S0  (1.0f / 16.0f)           // 1/sqrt(256)
#define INV_S1  (1.0f / 19.59591794f)    // 1/sqrt(384)
#define INV_S2  (1.0f / 17.88854382f)    // 1/sqrt(320)
#define INV_SL0 (1.0f / 26.53299832f)    // 1/sqrt(704)
#define INV_SL1 (1.0f / 11.31370850f)    // 1/sqrt(128)
#define INV_SL2 0.125f                   // 1/sqrt(64)

__device__ __forceinline__ v8f wmma_f32(v2f a, v2f b, v8f c) {
    // D = A(16x4,f32) * B(4x16,f32) + C(16x16,f32)
    return __builtin_amdgcn_wmma_f32_16x16x4_f32(
        false, a, false, b, (short)0, c, false, false);
}

__device__ __forceinline__ float sigf(float x) {
    return 1.0f / (1.0f + __expf(-x));
}

__global__ __launch_bounds__(256) void resblock_fused(
    const float* __restrict__ x,
    const float* __restrict__ W0a, const float* __restrict__ W1a, const float* __restrict__ W2a,
    const float* __restrict__ L0a, const float* __restrict__ L1a, const float* __restrict__ L2a,
    const float* __restrict__ b0a,
    const float* __restrict__ W0b, const float* __restrict__ W1b, const float* __restrict__ W2b,
    const float* __restrict__ L0b, const float* __restrict__ L1b, const float* __restrict__ L2b,
    const float* __restrict__ b0b,
    float* __restrict__ out, int nrows)
{
    extern __shared__ float sm[];
    const int tid    = threadIdx.x;
    const int lane   = tid & 31;
    const int wave   = tid >> 5;          // 0..7
    const int lane15 = lane & 15;
    const int hi     = lane >> 4;         // 0 or 1
    const int koff   = hi * 2;            // K sub-offset for A/B fragments
    const int r0     = blockIdx.x * 16;   // first node row of this tile

    // ---- load 16x960 input tile into LDS (coalesced float4) ----
    for (int i = tid; i < 3840; i += 256) {
        int m = i / 240, c4 = i % 240;
        int row = r0 + m;
        float4 v = make_float4(0.f, 0.f, 0.f, 0.f);
        if (row < nrows) v = *(const float4*)(x + (size_t)row * 960 + c4 * 4);
        *(float4*)(&sm[XB0 + m * XSTR + c4 * 4]) = v;
    }
    __syncthreads();

    for (int blk = 0; blk < 2; ++blk) {
        const float* W0 = blk ? W0b : W0a;
        const float* W1 = blk ? W1b : W1a;
        const float* W2 = blk ? W2b : W2a;
        const float* L0 = blk ? L0b : L0a;
        const float* L1 = blk ? L1b : L1a;
        const float* L2 = blk ? L2b : L2a;
        const float* b0 = blk ? b0b : b0a;
        float* xin  = sm + (blk ? XB1 : XB0);
        float* xout = sm + (blk ? XB0 : XB1);
        float* cat  = sm + CATO;
        float* gat  = sm + GATO;

        // ===== stage 1: d0 = s * (s @ W0^T) / 16  -> cat[:,0:256] =====
        for (int nt = wave * 2; nt < wave * 2 + 2; ++nt) {
            const int ncol = nt * 16 + lane15;
            const float* wrow = W0 + (size_t)ncol * 256;   // B[k][n] = W0[n][k]
            v8f acc = (v8f)0.0f;
            #pragma unroll 4
            for (int k0 = 0; k0 < 256; k0 += 4) {
                int k = k0 + koff;
                v2f a; a.x = xin[lane15 * XSTR + k];  a.y = xin[lane15 * XSTR + k + 1];
                v2f b; b.x = wrow[k];                 b.y = wrow[k + 1];
                acc = wmma_f32(a, b, acc);
            }
            #pragma unroll
            for (int r = 0; r < 8; ++r) {
                int m = r + 8 * hi;
                float sval = xin[m * XSTR + ncol];
                cat[m * CSTR + ncol] = sval * acc[r] * INV_S0;
            }
        }

        // ===== stage 2: d1 = sum_m v1 * (v1_m @ W1^T) / sqrt(384) -> cat[:,256:384] =====
        {
            const int u = wave * 16 + lane15;              // 8 waves x 16 = 128 cols
            const float* wrow = W1 + (size_t)u * 128;
            float d1a[8];
            #pragma unroll
            for (int r = 0; r < 8; ++r) d1a[r] = 0.0f;
            for (int mm = 0; mm < 3; ++mm) {
                v8f acc = (v8f)0.0f;
                #pragma unroll 4
                for (int k0 = 0; k0 < 128; k0 += 4) {
                    int k = k0 + koff;
                    v2f a; a.x = xin[lane15 * XSTR + 256 + 3 * k + mm];
                           a.y = xin[lane15 * XSTR + 256 + 3 * (k + 1) + mm];
                    v2f b; b.x = wrow[k]; b.y = wrow[k + 1];
                    acc = wmma_f32(a, b, acc);
                }
                #pragma unroll
                for (int r = 0; r < 8; ++r) {
                    int m = r + 8 * hi;
                    d1a[r] += xin[m * XSTR + 256 + 3 * u + mm] * acc[r];
                }
            }
            #pragma unroll
            for (int r = 0; r < 8; ++r) {
                int m = r + 8 * hi;
                cat[m * CSTR + 256 + u] = d1a[r] * INV_S1;
            }
        }

        // ===== stage 3: d2 -> cat[:,384:448]  (waves 0-3) ; copy s -> cat[:,448:704] (waves 4-7) =====
        if (wave < 4) {
            const int u = wave * 16 + lane15;              // 4 waves x 16 = 64 cols
            const float* wrow = W2 + (size_t)u * 64;
            float d2a[8];
            #pragma unroll
            for (int r = 0; r < 8; ++r) d2a[r] = 0.0f;
            for (int mm = 0; mm < 5; ++mm) {
                v8f acc = (v8f)0.0f;
                #pragma unroll 4
                for (int k0 = 0; k0 < 64; k0 += 4) {
                    int k = k0 + koff;
                    v2f a; a.x = xin[lane15 * XSTR + 640 + 5 * k + mm];
                           a.y = xin[lane15 * XSTR + 640 + 5 * (k + 1) + mm];
                    v2f b; b.x = wrow[k]; b.y = wrow[k + 1];
                    acc = wmma_f32(a, b, acc);
                }
                #pragma unroll
                for (int r = 0; r < 8; ++r) {
                    int m = r + 8 * hi;
                    d2a[r] += xin[m * XSTR + 640 + 5 * u + mm] * acc[r];
                }
            }
            #pragma unroll
            for (int r = 0; r < 8; ++r) {
                int m = r + 8 * hi;
                cat[m * CSTR + 384 + u] = d2a[r] * INV_S2;
            }
        } else {
            int t = tid - 128;
            for (int i = t; i < 4096; i += 128) {
                int m = i >> 8, c = i & 255;
                cat[m * CSTR + 448 + c] = xin[m * XSTR + c];
            }
        }
        __syncthreads();

        // ===== stage 5: h0 = cat @ L0 / sqrt(704) + b0; silu -> xout[:,0:256], gates -> gat =====
        for (int nt = wave; nt < 28; nt += 8) {
            const int n = nt * 16 + lane15;                // 0..447
            v8f acc = (v8f)0.0f;
            #pragma unroll 4
            for (int k0 = 0; k0 < 704; k0 += 4) {
                int k = k0 + koff;
                v2f a; a.x = cat[lane15 * CSTR + k];  a.y = cat[lane15 * CSTR + k + 1];
                v2f b; b.x = L0[(size_t)k * 448 + n]; b.y = L0[(size_t)(k + 1) * 448 + n];
                acc = wmma_f32(a, b, acc);
            }
            float bias = b0[n];
            #pragma unroll
            for (int r = 0; r < 8; ++r) {
                int m = r + 8 * hi;
                float h = acc[r] * INV_SL0 + bias;
                float sg = sigf(h);
                if (n < 256) xout[m * XSTR + n] = h * sg;          // silu
                else         gat[m * GSTR + (n - 256)] = sg;       // g1 | g2
            }
        }
        __syncthreads();

        // ===== stage 6: o1 = (v1_m @ L1)/sqrt(128) * g1 -> xout[:,256:640] =====
        {
            const int u = wave * 16 + lane15;              // 8 waves x 16 = 128 cols
            for (int mm = 0; mm < 3; ++mm) {
                v8f acc = (v8f)0.0f;
                #pragma unroll 4
                for (int k0 = 0; k0 < 128; k0 += 4) {
                    int k = k0 + koff;
                    v2f a; a.x = xin[lane15 * XSTR + 256 + 3 * k + mm];
                           a.y = xin[lane15 * XSTR + 256 + 3 * (k + 1) + mm];
                    v2f b; b.x = L1[(size_t)k * 128 + u]; b.y = L1[(size_t)(k + 1) * 128 + u];
                    acc = wmma_f32(a, b, acc);
                }
                #pragma unroll
                for (int r = 0; r < 8; ++r) {
                    int m = r + 8 * hi;
                    xout[m * XSTR + 256 + 3 * u + mm] =
                        acc[r] * INV_SL1 * gat[m * GSTR + u];
                }
            }
        }

        // ===== stage 7: o2 = (v2_m @ L2)/sqrt(64) * g2 -> xout[:,640:960] =====
        for (int j = wave; j < 20; j += 8) {
            const int nt = j & 3, mm = j >> 2;
            const int u = nt * 16 + lane15;
            v8f acc = (v8f)0.0f;
            #pragma unroll 4
            for (int k0 = 0; k0 < 64; k0 += 4) {
                int k = k0 + koff;
                v2f a; a.x = xin[lane15 * XSTR + 640 + 5 * k + mm];
                       a.y = xin[lane15 * XSTR + 640 + 5 * (k + 1) + mm];
                v2f b; b.x = L2[(size_t)k * 64 + u]; b.y = L2[(size_t)(k + 1) * 64 + u];
                acc = wmma_f32(a, b, acc);
            }
            #pragma unroll
            for (int r = 0; r < 8; ++r) {
                int m = r + 8 * hi;
                xout[m * XSTR + 640 + 5 * u + mm] =
                    acc[r] * INV_SL2 * gat[m * GSTR + 128 + u];
            }
        }
        __syncthreads();
    }

    // ---- residual + store: out = xbuf0 + node_input ----
    for (int i = tid; i < 3840; i += 256) {
        int m = i / 240, c4 = i % 240;
        int row = r0 + m;
        if (row < nrows) {
            const float4 xv = *(const float4*)(x + (size_t)row * 960 + c4 * 4);
            float4 hv = *(const float4*)(&sm[XB0 + m * XSTR + c4 * 4]);
            float4 o;
            o.x = hv.x + xv.x; o.y = hv.y + xv.y;
            o.z = hv.z + xv.z; o.w = hv.w + xv.w;
            *(float4*)(out + (size_t)row * 960 + c4 * 4) = o;
        }
    }
}

extern "C" void kernel_launch(void* const* d_in, const int* in_sizes, int n_in,
                              void* d_out, int out_size, void* d_ws, size_t ws_size,
                              hipStream_t stream) {
    // setup_inputs() dict order:
    // 0: node_input, 1: W0_1, 2: W1_1, 3: W2_1, 4: L0_1, 5: L1_1, 6: L2_1, 7: b0_1,
    // 8: W0_2, 9: W1_2, 10: W2_2, 11: L0_2, 12: L1_2, 13: L2_2, 14: b0_2
    const float* x   = (const float*)d_in[0];
    const float* W0a = (const float*)d_in[1];
    const float* W1a = (const float*)d_in[2];
    const float* W2a = (const float*)d_in[3];
    const float* L0a = (const float*)d_in[4];
    const float* L1a = (const float*)d_in[5];
    const float* L2a = (const float*)d_in[6];
    const float* b0a = (const float*)d_in[7];
    const float* W0b = (const float*)d_in[8];
    const float* W1b = (const float*)d_in[9];
    const float* W2b = (const float*)d_in[10];
    const float* L0b = (const float*)d_in[11];
    const float* L1b = (const float*)d_in[12];
    const float* L2b = (const float*)d_in[13];
    const float* b0b = (const float*)d_in[14];

    const int nrows = in_sizes[0] / 960;
    const int grid  = (nrows + 15) / 16;
    const size_t shmem = (size_t)SMEM_FLOATS * sizeof(float);

    resblock_fused<<<grid, 256, shmem, stream>>>(
        x, W0a, W1a, W2a, L0a, L1a, L2a, b0a,
           W0b, W1b, W2b, L0b, L1b, L2b, b0b,
        (float*)d_out, nrows);
}